// Basic3DBlock_54889682043712
// MI455X (gfx1250) — compile-verified
//
#include <hip/hip_runtime.h>

typedef float v2f __attribute__((ext_vector_type(2)));
typedef float v8f __attribute__((ext_vector_type(8)));

constexpr int   NVOX  = 2000000;       // active voxels
constexpr int   TAPS  = 27;
constexpr int   PAIRS = NVOX / 32;     // 62500 exact: each wave iter = 2 tiles (32 voxels)
constexpr int   NB    = 1024;          // conv grid blocks (8192 waves)
constexpr float EPS   = 1e-5f;

// d_ws float layout:
//  [0..15]                zero feature row (for missing neighbors)
//  [16..31]               per-channel scale
//  [32..47]               per-channel shift
//  [48 .. 48+NB*32)       per-block partials: sum[16], sumsq[16]

__global__ void init_ws(float* ws) {
  if (threadIdx.x < 16) ws[threadIdx.x] = 0.0f;
}

__global__ __launch_bounds__(256)
void conv_wmma(const float* __restrict__ feat,
               const float* __restrict__ wgt,
               const int*   __restrict__ nbr,
               float*       __restrict__ out,
               float*       __restrict__ ws) {
  __shared__ float s_wgt[TAPS * 256];   // 27 x 16x16 fp32 weights (27.6 KB)
  __shared__ float s_red[8][32];        // per-wave channel partials

  const int tid = threadIdx.x;
  // stage all 27 weight matrices into LDS as 128-bit chunks
  {
    const float4* src = (const float4*)wgt;
    float4*       dst = (float4*)s_wgt;
    for (int i = tid; i < TAPS * 64; i += 256) dst[i] = src[i];
  }
  __syncthreads();

  const int lane = tid & 31;
  const int wv   = tid >> 5;
  const int half = lane >> 4;   // half-wave: K split for A/B fragments
  const int m    = lane & 15;   // M row (A) / N column (B,C,D)

  float sacc = 0.0f, qacc = 0.0f;

  const int gw = blockIdx.x * 8 + wv;   // global wave id
  const int nw = NB * 8;                // total waves

  for (int p = gw; p < PAIRS; p += nw) {
    const int base  = p * 32;
    const int pnext = (base + nw * 32 < NVOX) ? (base + nw * 32) : base;

    v8f c0 = {};   // accumulator, tile 0 (voxels base..base+15)
    v8f c1 = {};   // accumulator, tile 1 (voxels base+16..base+31)

    for (int k = 0; k < TAPS; ++k) {
      const int* irow = nbr + k * NVOX;
      // prefetch next pair's 128B index line for this tap (streaming 216 MB)
      __builtin_prefetch(irow + pnext + lane, 0, 1);

      const int idx0 = irow[base + m];
      const int idx1 = irow[base + 16 + m];
      // missing neighbor (idx == NVOX) -> zero row held in ws
      const float* rp0 = (idx0 < NVOX) ? (feat + idx0 * 16 + half * 2)
                                       : (ws + half * 2);
      const float* rp1 = (idx1 < NVOX) ? (feat + idx1 * 16 + half * 2)
                                       : (ws + half * 2);

      // A: 16x16 fp32 tile as four 16x4 slices (ISA 7.12.2):
      //   lanes 0-15 : M=lane,    K = 4s+{0,1}
      //   lanes 16-31: M=lane-16, K = 4s+{2,3}
      #pragma unroll
      for (int s = 0; s < 4; ++s) {
        v2f a0, a1, b;
        a0.x = rp0[4 * s + 0];
        a0.y = rp0[4 * s + 1];
        a1.x = rp1[4 * s + 0];
        a1.y = rp1[4 * s + 1];
        // B: 4x16 slice of the 16x16 weights, column m, rows K = 4s+half*2+{0,1}
        const int off = k * 256 + 64 * s + 32 * half + m;
        b.x = s_wgt[off];
        b.y = s_wgt[off + 16];
        // two independent C chains -> interleaved WMMAs hide the f32 RAW latency
        c0 = __builtin_amdgcn_wmma_f32_16x16x4_f32(
                 false, a0, false, b, (short)0, c0, false, false);
        c1 = __builtin_amdgcn_wmma_f32_16x16x4_f32(
                 false, a1, false, b, (short)0, c1, false, false);
      }
    }

    // C/D layout: VGPR r -> row (r + 8*half), column m. Store + accumulate stats.
    #pragma unroll
    for (int r = 0; r < 8; ++r) {
      const float v0 = c0[r];
      const float v1 = c1[r];
      out[(base + r + 8 * half) * 16 + m]      = v0;
      out[(base + 16 + r + 8 * half) * 16 + m] = v1;
      sacc += v0 + v1;
      qacc += v0 * v0 + v1 * v1;
    }
  }

  // combine the two half-waves (same channel m, different rows)
  sacc += __shfl_xor(sacc, 16, 32);
  qacc += __shfl_xor(qacc, 16, 32);
  if (half == 0) {
    s_red[wv][m]      = sacc;
    s_red[wv][16 + m] = qacc;
  }
  __syncthreads();

  // deterministic per-block partial: t<16 -> sum, t in 16..31 -> sumsq
  if (tid < 32) {
    float acc = 0.0f;
    #pragma unroll
    for (int w = 0; w < 8; ++w) acc += s_red[w][tid];
    ws[48 + blockIdx.x * 32 + tid] = acc;
  }
}

__global__ void bn_stats(float* __restrict__ ws,
                         const float* __restrict__ gamma,
                         const float* __restrict__ beta) {
  __shared__ float red[32];
  const int t = threadIdx.x;   // 32 threads
  float acc = 0.0f;
  for (int b = 0; b < NB; ++b) acc += ws[48 + b * 32 + t];
  red[t] = acc;
  __syncthreads();
  if (t < 16) {
    const float invn = 1.0f / (float)NVOX;
    const float mean = red[t] * invn;
    const float var  = red[16 + t] * invn - mean * mean;
    const float sc   = gamma[t] * rsqrtf(var + EPS);
    ws[16 + t] = sc;
    ws[32 + t] = beta[t] - mean * sc;
  }
}

__global__ __launch_bounds__(256)
void bn_relu(float* __restrict__ out, const float* __restrict__ ws) {
  const int e  = (blockIdx.x * 256 + threadIdx.x) * 4;  // < 32M, fits int
  const int c0 = e & 15;                                // channel of first elem
  float4 v  = *(const float4*)(out + e);
  float4 sc = *(const float4*)(ws + 16 + c0);
  float4 sh = *(const float4*)(ws + 32 + c0);
  float4 r;
  r.x = fmaxf(fmaf(v.x, sc.x, sh.x), 0.0f);
  r.y = fmaxf(fmaf(v.y, sc.y, sh.y), 0.0f);
  r.z = fmaxf(fmaf(v.z, sc.z, sh.z), 0.0f);
  r.w = fmaxf(fmaf(v.w, sc.w, sh.w), 0.0f);
  *(float4*)(out + e) = r;
}

extern "C" void kernel_launch(void* const* d_in, const int* in_sizes, int n_in,
                              void* d_out, int out_size, void* d_ws, size_t ws_size,
                              hipStream_t stream) {
  const float* feat  = (const float*)d_in[0];   // [N,16]
  const float* wgt   = (const float*)d_in[1];   // [27,16,16]
  const float* gamma = (const float*)d_in[2];   // [16]
  const float* beta  = (const float*)d_in[3];   // [16]
  const int*   nbr   = (const int*)d_in[4];     // [27,N]
  float* out = (float*)d_out;
  float* ws  = (float*)d_ws;

  init_ws<<<1, 16, 0, stream>>>(ws);
  conv_wmma<<<NB, 256, 0, stream>>>(feat, wgt, nbr, out, ws);
  bn_stats<<<1, 32, 0, stream>>>(ws, gamma, beta);
  bn_relu<<<(NVOX * 16 / 4) / 256, 256, 0, stream>>>(out, ws);
}